// ConvClassifier_3143916060932
// MI455X (gfx1250) — compile-verified
//
#include <hip/hip_runtime.h>

// ---------------------------------------------------------------------------
// Predictive-coding network (20 steps) on MI455X / gfx1250.
//
//  * Batch (2048) lives in the GEMM N dimension so weights (W2/W3/W4, L2-
//    resident) are amortized across thousands of columns; per-sample
//    activation slabs are staged in the 320KB WGP LDS, weight K-panels are
//    staged in LDS and shared by all 8 waves of the workgroup.
//  * sumpool2(conv5x5(.)) is folded to a stride-2 6x6 conv with pooled
//    kernels K2/K3 precomputed once into the workspace.
//  * All matrix math: V_WMMA_F32_16X16X4_F32 (wave32). fp32 keeps the
//    iterative dynamics faithful; workload is HBM-bound (~0.8GB/step at
//    23.3TB/s), not matrix-rate bound.
//
// Fragment conventions (ISA 7.12.2):
//   A(16x4):  lane&15 = M, K = 2*(lane>>4) + vgpr
//   B(4x16):  lane&15 = N, K = 2*(lane>>4) + vgpr
//   D(16x16): vgpr r -> M = r + 8*(lane>>4), N = lane&15
// ---------------------------------------------------------------------------

typedef float v2f __attribute__((ext_vector_type(2)));
typedef float v8f __attribute__((ext_vector_type(8)));

#define GAMMA 0.1f
#define STEPS 20

__device__ __forceinline__ v8f vzero() {
  v8f v = {0.f, 0.f, 0.f, 0.f, 0.f, 0.f, 0.f, 0.f};
  return v;
}

__device__ __forceinline__ v8f wmma4(float a0, float a1, float b0, float b1, v8f c) {
  v2f a = {a0, a1};
  v2f b = {b0, b1};
  return __builtin_amdgcn_wmma_f32_16x16x4_f32(false, a, false, b, (short)0, c,
                                               false, false);
}

// ---------------------------------------------------------------------------
// Init: X* = 0.01 * inputs; build pooled kernels
//   K2[oc][ (a6*6+b6)*32 + ic ] = sum_{dy,dx in {0,1}, in-range} W2[oc,ic,a6-dy,b6-dx]
//   K3 analogous from W3.
// ---------------------------------------------------------------------------
__global__ __launch_bounds__(256) void k_init(
    const float* __restrict__ x1i, const float* __restrict__ x2i,
    const float* __restrict__ x3i, const float* __restrict__ x4i,
    const float* __restrict__ x5i, const float* __restrict__ W2g,
    const float* __restrict__ W3g, float* __restrict__ X1, float* __restrict__ X2,
    float* __restrict__ X3, float* __restrict__ X4, float* __restrict__ X5,
    float* __restrict__ K2, float* __restrict__ K3, int B) {
  const long long tid = (long long)blockIdx.x * 256 + threadIdx.x;
  const long long stride = (long long)gridDim.x * 256;
  for (long long i = tid; i < (long long)B * 18432; i += stride) X1[i] = 0.01f * x1i[i];
  for (long long i = tid; i < (long long)B * 6400; i += stride) X2[i] = 0.01f * x2i[i];
  for (long long i = tid; i < (long long)B * 576; i += stride) X3[i] = 0.01f * x3i[i];
  for (long long i = tid; i < (long long)B * 128; i += stride) X4[i] = 0.01f * x4i[i];
  for (long long i = tid; i < (long long)B * 10; i += stride) X5[i] = 0.01f * x5i[i];
  for (long long t = tid; t < 64 * 1152; t += stride) {
    int m = (int)(t / 1152), k = (int)(t % 1152);
    int ic = k & 31, tap = k >> 5, a6 = tap / 6, b6 = tap % 6;
    float s = 0.f;
    for (int dy = 0; dy < 2; ++dy) {
      int a = a6 - dy;
      if (a < 0 || a > 4) continue;
      for (int dx = 0; dx < 2; ++dx) {
        int bb = b6 - dx;
        if (bb < 0 || bb > 4) continue;
        s += W2g[m * 800 + ic * 25 + a * 5 + bb];
      }
    }
    K2[t] = s;
  }
  for (long long t = tid; t < 64 * 2304; t += stride) {
    int m = (int)(t / 2304), k = (int)(t % 2304);
    int ic = k & 63, tap = k >> 6, a6 = tap / 6, b6 = tap % 6;
    float s = 0.f;
    for (int dy = 0; dy < 2; ++dy) {
      int a = a6 - dy;
      if (a < 0 || a > 4) continue;
      for (int dx = 0; dx < 2; ++dx) {
        int bb = b6 - dx;
        if (bb < 0 || bb > 4) continue;
        s += W3g[m * 1600 + ic * 25 + a * 5 + bb];
      }
    }
    K3[t] = s;
  }
}

// ---------------------------------------------------------------------------
// e0 (VALU, per-sample, x1 slab staged in LDS) + e4 (tiny FC) in one kernel.
//   e0[b,p] = obs - (b1 + sum_{ic,a,b} W1[ic,0,a,b] * relu(x1[ic, y-a, x-b]))
//   e4[b,i] = x4 - (b5[i] + sum_j relu(x5[j]) * W5[i,j])
// ---------------------------------------------------------------------------
__global__ __launch_bounds__(256) void k_e0e4(
    const float* __restrict__ obs, const float* __restrict__ W1g,
    const float* __restrict__ b1g, const float* __restrict__ W5g,
    const float* __restrict__ b5g, const float* __restrict__ X1,
    const float* __restrict__ X4, const float* __restrict__ X5,
    float* __restrict__ E0, float* __restrict__ E4, int B) {
  __shared__ float sm[19232];  // [0,18432): relu(x1 sample); [18432,19232): W1
  const int tid = threadIdx.x;
  if ((int)blockIdx.x < B) {
    const int b = blockIdx.x;
    const float* x1g = X1 + b * 18432;
    for (int i = tid; i < 18432; i += 256) sm[i] = fmaxf(x1g[i], 0.0f);
    for (int i = tid; i < 800; i += 256) sm[18432 + i] = W1g[i];
    __syncthreads();
    const float bias = b1g[0];
    for (int p = tid; p < 784; p += 256) {
      const int y = p / 28, x = p % 28;
      float acc = bias;
      for (int a = 0; a < 5; ++a) {
        const int i = y - a;
        if (i < 0 || i >= 24) continue;
        for (int bb = 0; bb < 5; ++bb) {
          const int j = x - bb;
          if (j < 0 || j >= 24) continue;
          const float* w = &sm[18432 + a * 5 + bb];
          const float* xs = &sm[i * 24 + j];
          float s = 0.f;
#pragma unroll 8
          for (int ic = 0; ic < 32; ++ic) s += w[ic * 25] * xs[ic * 576];
          acc += s;
        }
      }
      E0[b * 784 + p] = obs[b * 784 + p] - acc;
    }
  } else {
    const int g = ((int)blockIdx.x - B) * 256 + tid;
    if (g < B * 128) {
      const int bs = g >> 7, i = g & 127;
      float acc = b5g[i];
#pragma unroll
      for (int j = 0; j < 10; ++j)
        acc += fmaxf(X5[bs * 10 + j], 0.0f) * W5g[i * 10 + j];
      E4[g] = X4[g] - acc;
    }
  }
}

// ---------------------------------------------------------------------------
// e1 = x1 - (convT2(up2(relu(x2)), W2) + b2)   [WMMA, one workgroup = sample]
// GEMM: M=32(oc), N=576(pix), K=1600=(tap 25)x(ic 64)
//   A[m,k] = W2[ic*800 + m*25 + tap]   (staged tap-panel 32x64 in LDS)
//   B[k,n] = relu(x2)[ic, (py-a)>>1, (px-b)>>1] with spatial mask
// ---------------------------------------------------------------------------
__global__ __launch_bounds__(256) void k_e1(
    const float* __restrict__ W2g, const float* __restrict__ b2g,
    const float* __restrict__ X1, const float* __restrict__ X2,
    float* __restrict__ E1) {
  __shared__ float x2s[6400];
  __shared__ float pa[2048];
  __shared__ float b2s[32];
  const int b = blockIdx.x, tid = threadIdx.x;
  const int wave = tid >> 5, lane = tid & 31, lh = lane >> 4, lm = lane & 15;

  const float* x2g = X2 + b * 6400;
  for (int i = tid; i < 6400; i += 256) x2s[i] = fmaxf(x2g[i], 0.0f);
  if (tid < 32) b2s[tid] = b2g[tid];

  int ntl[5];
  int ntn = 0;
  for (int t = wave; t < 36; t += 8) ntl[ntn++] = t;
  int py[5], px[5];
#pragma unroll
  for (int t = 0; t < 5; ++t) {
    const int n = ((t < ntn) ? ntl[t] : 0) * 16 + lm;
    py[t] = n / 24;
    px[t] = n % 24;
  }
  v8f acc[5][2];
#pragma unroll
  for (int t = 0; t < 5; ++t) {
    acc[t][0] = vzero();
    acc[t][1] = vzero();
  }

  for (int tap = 0; tap < 25; ++tap) {
    const int a = tap / 5, bb = tap % 5;
    __syncthreads();
    for (int i = tid; i < 2048; i += 256) {
      const int m = i >> 6, ic = i & 63;
      pa[i] = W2g[ic * 800 + m * 25 + tap];
    }
    __syncthreads();
    int ub[5];
    float msk[5];
#pragma unroll
    for (int t = 0; t < 5; ++t) {
      const int ii = py[t] - a, jj = px[t] - bb;
      const bool v = (ii >= 0) && (ii < 20) && (jj >= 0) && (jj < 20);
      ub[t] = v ? ((ii >> 1) * 10 + (jj >> 1)) : 0;
      msk[t] = v ? 1.0f : 0.0f;
    }
    for (int icb = 0; icb < 64; icb += 4) {
      const int ick = icb + lh * 2;
      const float a00 = pa[lm * 64 + ick], a01 = pa[lm * 64 + ick + 1];
      const float a10 = pa[(16 + lm) * 64 + ick], a11 = pa[(16 + lm) * 64 + ick + 1];
#pragma unroll
      for (int t = 0; t < 5; ++t) {
        const float bv0 = msk[t] * x2s[ick * 100 + ub[t]];
        const float bv1 = msk[t] * x2s[(ick + 1) * 100 + ub[t]];
        acc[t][0] = wmma4(a00, a01, bv0, bv1, acc[t][0]);
        acc[t][1] = wmma4(a10, a11, bv0, bv1, acc[t][1]);
      }
    }
  }
  for (int t = 0; t < ntn; ++t) {
#pragma unroll
    for (int mt = 0; mt < 2; ++mt) {
#pragma unroll
      for (int r = 0; r < 8; ++r) {
        const int row = mt * 16 + r + 8 * lh;
        const int col = ntl[t] * 16 + lm;
        const int idx = b * 18432 + row * 576 + col;
        E1[idx] = X1[idx] - (acc[t][mt][r] + b2s[row]);
      }
    }
  }
}

// ---------------------------------------------------------------------------
// e2 = x2 - (convT3(up2(relu(x3)), W3) + b3)   [WMMA, one workgroup = sample]
// GEMM: M=64, N=100, K=1600=(tap 25)x(ic 64)
// ---------------------------------------------------------------------------
__global__ __launch_bounds__(256) void k_e2(
    const float* __restrict__ W3g, const float* __restrict__ b3g,
    const float* __restrict__ X2, const float* __restrict__ X3,
    float* __restrict__ E2) {
  __shared__ float x3s[576];
  __shared__ float pa[4096];
  __shared__ float b3s[64];
  const int b = blockIdx.x, tid = threadIdx.x;
  const int wave = tid >> 5, lane = tid & 31, lh = lane >> 4, lm = lane & 15;

  for (int i = tid; i < 576; i += 256) x3s[i] = fmaxf(X3[b * 576 + i], 0.0f);
  if (tid < 64) b3s[tid] = b3g[tid];

  const int n = wave * 16 + lm;  // wave 7 computes a fully-masked dummy tile
  const bool nv = n < 100;
  const int py = n / 10, px = n % 10;
  v8f acc[4];
#pragma unroll
  for (int mt = 0; mt < 4; ++mt) acc[mt] = vzero();

  for (int tap = 0; tap < 25; ++tap) {
    const int a = tap / 5, bb = tap % 5;
    __syncthreads();
    for (int i = tid; i < 4096; i += 256) {
      const int m = i >> 6, ic = i & 63;
      pa[i] = W3g[ic * 1600 + m * 25 + tap];
    }
    __syncthreads();
    const int ii = py - a, jj = px - bb;
    const bool v = nv && (ii >= 0) && (ii < 6) && (jj >= 0) && (jj < 6);
    const int ub = v ? ((ii >> 1) * 3 + (jj >> 1)) : 0;
    const float msk = v ? 1.0f : 0.0f;
    for (int icb = 0; icb < 64; icb += 4) {
      const int ick = icb + lh * 2;
      const float bv0 = msk * x3s[ick * 9 + ub];
      const float bv1 = msk * x3s[(ick + 1) * 9 + ub];
#pragma unroll
      for (int mt = 0; mt < 4; ++mt) {
        const float a0 = pa[(mt * 16 + lm) * 64 + ick];
        const float a1 = pa[(mt * 16 + lm) * 64 + ick + 1];
        acc[mt] = wmma4(a0, a1, bv0, bv1, acc[mt]);
      }
    }
  }
  if (nv) {
#pragma unroll
    for (int mt = 0; mt < 4; ++mt) {
#pragma unroll
      for (int r = 0; r < 8; ++r) {
        const int row = mt * 16 + r + 8 * lh;
        const int idx = b * 6400 + row * 100 + n;
        E2[idx] = X2[idx] - (acc[mt][r] + b3s[row]);
      }
    }
  }
}

// ---------------------------------------------------------------------------
// e3 = x3 - (relu(x4) @ W4^T + b4)   [WMMA, workgroup = 16 samples]
// GEMM: M=576(i), N=16(batch), K=128;  A[m,k]=W4[m*128+k], B staged relu(x4)
// ---------------------------------------------------------------------------
__global__ __launch_bounds__(256) void k_e3(
    const float* __restrict__ W4g, const float* __restrict__ b4g,
    const float* __restrict__ X3, const float* __restrict__ X4,
    float* __restrict__ E3) {
  __shared__ float x4s[2048];  // 16 samples x 128, relu applied
  const int sb = blockIdx.x * 16, tid = threadIdx.x;
  const int wave = tid >> 5, lane = tid & 31, lh = lane >> 4, lm = lane & 15;
  for (int i = tid; i < 2048; i += 256) x4s[i] = fmaxf(X4[sb * 128 + i], 0.0f);
  __syncthreads();

  int mtl[5];
  bool mv[5];
#pragma unroll
  for (int s = 0; s < 5; ++s) {
    const int mt = wave + 8 * s;
    mv[s] = mt < 36;
    mtl[s] = mv[s] ? mt : wave;
  }
  v8f acc[5];
#pragma unroll
  for (int s = 0; s < 5; ++s) acc[s] = vzero();

  for (int kc = 0; kc < 128; kc += 4) {
    const int kk = kc + lh * 2;
    const float bv0 = x4s[lm * 128 + kk];
    const float bv1 = x4s[lm * 128 + kk + 1];
#pragma unroll
    for (int s = 0; s < 5; ++s) {
      const int m = mtl[s] * 16 + lm;
      const float a0 = W4g[m * 128 + kk];
      const float a1 = W4g[m * 128 + kk + 1];
      acc[s] = wmma4(a0, a1, bv0, bv1, acc[s]);
    }
  }
#pragma unroll
  for (int s = 0; s < 5; ++s) {
    if (!mv[s]) continue;
#pragma unroll
    for (int r = 0; r < 8; ++r) {
      const int row = mtl[s] * 16 + r + 8 * lh;
      const int idx = (sb + lm) * 576 + row;
      E3[idx] = X3[idx] - (acc[s][r] + b4g[row]);
    }
  }
}

// ---------------------------------------------------------------------------
// x1 update (VALU conv) + x5 update (tiny FC) in one kernel.
//   x1 += g*(conv(e0,W1)*drelu(x1) - e1);  x5 += g*((e4@W5)*drelu(x5))
// ---------------------------------------------------------------------------
__global__ __launch_bounds__(256) void k_x1x5(
    const float* __restrict__ W1g, const float* __restrict__ W5g,
    const float* __restrict__ E0, const float* __restrict__ E1,
    const float* __restrict__ E4, float* __restrict__ X1,
    float* __restrict__ X5, int B) {
  __shared__ float sm[1584];  // e0 (784) + W1 (800)
  const int tid = threadIdx.x;
  if ((int)blockIdx.x < B) {
    const int b = blockIdx.x;
    for (int i = tid; i < 784; i += 256) sm[i] = E0[b * 784 + i];
    for (int i = tid; i < 800; i += 256) sm[784 + i] = W1g[i];
    __syncthreads();
    for (int idx = tid; idx < 18432; idx += 256) {
      const int c = idx / 576, rem = idx % 576, y = rem / 24, x = rem % 24;
      const float* w = &sm[784 + c * 25];
      const float* e = &sm[y * 28 + x];
      float g = 0.f;
#pragma unroll
      for (int a = 0; a < 5; ++a)
#pragma unroll
        for (int bb = 0; bb < 5; ++bb) g += w[a * 5 + bb] * e[a * 28 + bb];
      const int gi = b * 18432 + idx;
      const float xv = X1[gi];
      X1[gi] = xv + GAMMA * (g * (xv > 0.0f ? 1.0f : 0.0f) - E1[gi]);
    }
  } else {
    const int g = ((int)blockIdx.x - B) * 256 + tid;
    if (g < B * 10) {
      const int bs = g / 10, j = g % 10;
      float s = 0.f;
#pragma unroll 16
      for (int i = 0; i < 128; ++i) s += E4[bs * 128 + i] * W5g[i * 10 + j];
      const float xv = X5[g];
      X5[g] = xv + GAMMA * s * (xv > 0.0f ? 1.0f : 0.0f);
    }
  }
}

// ---------------------------------------------------------------------------
// x2 update [WMMA, workgroup = sample]: s = pooled-conv(e1, K2), stride-2 6x6
// GEMM: M=64, N=100, K=1152=(tap 36)x(ic 32); e1 slab (73.7KB) staged in LDS.
// ---------------------------------------------------------------------------
__global__ __launch_bounds__(256) void k_x2(
    const float* __restrict__ K2, const float* __restrict__ E1,
    const float* __restrict__ E2, float* __restrict__ X2) {
  __shared__ float e1s[18432];
  __shared__ float pk[2048];
  const int b = blockIdx.x, tid = threadIdx.x;
  const int wave = tid >> 5, lane = tid & 31, lh = lane >> 4, lm = lane & 15;
  for (int i = tid; i < 18432; i += 256) e1s[i] = E1[b * 18432 + i];

  const int n = wave * 16 + lm;  // wave 7 = masked dummy tile
  const bool nv = n < 100;
  const int Y = n / 10, Xc = n % 10;
  v8f acc[4];
#pragma unroll
  for (int mt = 0; mt < 4; ++mt) acc[mt] = vzero();

  for (int tap = 0; tap < 36; ++tap) {
    const int a6 = tap / 6, b6 = tap % 6;
    __syncthreads();
    for (int i = tid; i < 2048; i += 256) {
      const int m = i >> 5, ic = i & 31;
      pk[i] = K2[m * 1152 + tap * 32 + ic];
    }
    __syncthreads();
    const int base = nv ? ((2 * Y + a6) * 24 + (2 * Xc + b6)) : 0;
    const float msk = nv ? 1.0f : 0.0f;
    for (int icb = 0; icb < 32; icb += 4) {
      const int ick = icb + lh * 2;
      const float bv0 = msk * e1s[ick * 576 + base];
      const float bv1 = msk * e1s[(ick + 1) * 576 + base];
#pragma unroll
      for (int mt = 0; mt < 4; ++mt) {
        const float a0 = pk[(mt * 16 + lm) * 32 + ick];
        const float a1 = pk[(mt * 16 + lm) * 32 + ick + 1];
        acc[mt] = wmma4(a0, a1, bv0, bv1, acc[mt]);
      }
    }
  }
  if (nv) {
#pragma unroll
    for (int mt = 0; mt < 4; ++mt) {
#pragma unroll
      for (int r = 0; r < 8; ++r) {
        const int row = mt * 16 + r + 8 * lh;
        const int idx = b * 6400 + row * 100 + n;
        const float xv = X2[idx];
        X2[idx] = xv + GAMMA * (acc[mt][r] * (xv > 0.0f ? 1.0f : 0.0f) - E2[idx]);
      }
    }
  }
}

// ---------------------------------------------------------------------------
// x3 update [WMMA, workgroup = 4 samples]: pooled-conv(e2, K3), stride-2 6x6
// GEMM: M=64, N=36(4 samples x 9 pix), K=2304=(tap 36)x(ic 64)
// ---------------------------------------------------------------------------
__global__ __launch_bounds__(256) void k_x3(
    const float* __restrict__ K3, const float* __restrict__ E2,
    const float* __restrict__ E3, float* __restrict__ X3) {
  __shared__ float e2s[25600];
  __shared__ float pk3[4096];
  const int sb = blockIdx.x * 4, tid = threadIdx.x;
  const int wave = tid >> 5, lane = tid & 31, lh = lane >> 4, lm = lane & 15;
  for (int i = tid; i < 25600; i += 256) e2s[i] = E2[sb * 6400 + i];

  // 12 (m-tile, n-tile) pairs over 8 waves; slot 1 real only for waves 0-3
  int pr[2];
  pr[0] = wave;
  const bool s1v = (wave + 8) < 12;
  pr[1] = s1v ? (wave + 8) : wave;
  int mtile[2], ncol[2], nbase9[2], YY[2], XX[2];
  bool pv[2];
#pragma unroll
  for (int s = 0; s < 2; ++s) {
    const int mt = pr[s] / 3, nt = pr[s] % 3;
    mtile[s] = mt;
    const int nn = nt * 16 + lm;
    ncol[s] = nn;
    pv[s] = nn < 36;
    const int sl = pv[s] ? (nn / 9) : 0;
    const int pp = pv[s] ? (nn % 9) : 0;
    nbase9[s] = sl * 6400;
    YY[s] = pp / 3;
    XX[s] = pp % 3;
  }
  v8f acc[2];
  acc[0] = vzero();
  acc[1] = vzero();

  for (int tap = 0; tap < 36; ++tap) {
    const int a6 = tap / 6, b6 = tap % 6;
    __syncthreads();
    for (int i = tid; i < 4096; i += 256) {
      const int m = i >> 6, ic = i & 63;
      pk3[i] = K3[m * 2304 + tap * 64 + ic];
    }
    __syncthreads();
#pragma unroll
    for (int s = 0; s < 2; ++s) {
      const int base = nbase9[s] + (2 * YY[s] + a6) * 10 + (2 * XX[s] + b6);
      const float msk = pv[s] ? 1.0f : 0.0f;
      for (int icb = 0; icb < 64; icb += 4) {
        const int ick = icb + lh * 2;
        const float bv0 = msk * e2s[base + ick * 100];
        const float bv1 = msk * e2s[base + (ick + 1) * 100];
        const float a0 = pk3[(mtile[s] * 16 + lm) * 64 + ick];
        const float a1 = pk3[(mtile[s] * 16 + lm) * 64 + ick + 1];
        acc[s] = wmma4(a0, a1, bv0, bv1, acc[s]);
      }
    }
  }
#pragma unroll
  for (int s = 0; s < 2; ++s) {
    if (s == 1 && !s1v) continue;
    if (!pv[s]) continue;
    const int sample = sb + ncol[s] / 9;
    const int pp = ncol[s] % 9;
#pragma unroll
    for (int r = 0; r < 8; ++r) {
      const int row = mtile[s] * 16 + r + 8 * lh;  // channel
      const int idx = sample * 576 + row * 9 + pp;
      const float xv = X3[idx];
      X3[idx] = xv + GAMMA * (acc[s][r] * (xv > 0.0f ? 1.0f : 0.0f) - E3[idx]);
    }
  }
}

// ---------------------------------------------------------------------------
// x4 update [WMMA, workgroup = 16 samples]: g = e3 @ W4
// GEMM: M=128(j), N=16(batch), K=576;  A[m,k]=W4[k*128+m]; e3 slab staged.
// ---------------------------------------------------------------------------
__global__ __launch_bounds__(256) void k_x4(
    const float* __restrict__ W4g, const float* __restrict__ E3,
    const float* __restrict__ E4, float* __restrict__ X4) {
  __shared__ float e3s[9216];  // 16 samples x 576
  const int sb = blockIdx.x * 16, tid = threadIdx.x;
  const int wave = tid >> 5, lane = tid & 31, lh = lane >> 4, lm = lane & 15;
  for (int i = tid; i < 9216; i += 256) e3s[i] = E3[sb * 576 + i];
  __syncthreads();

  v8f acc = vzero();
  const int m = wave * 16 + lm;  // 8 waves x 16 = 128 rows exactly
  for (int kc = 0; kc < 576; kc += 4) {
    const int kk = kc + lh * 2;
    const float bv0 = e3s[lm * 576 + kk];
    const float bv1 = e3s[lm * 576 + kk + 1];
    const float a0 = W4g[kk * 128 + m];
    const float a1 = W4g[(kk + 1) * 128 + m];
    acc = wmma4(a0, a1, bv0, bv1, acc);
  }
#pragma unroll
  for (int r = 0; r < 8; ++r) {
    const int row = wave * 16 + r + 8 * lh;
    const int idx = (sb + lm) * 128 + row;
    const float xv = X4[idx];
    X4[idx] = xv + GAMMA * (acc[r] * (xv > 0.0f ? 1.0f : 0.0f) - E4[idx]);
  }
}

__global__ __launch_bounds__(256) void k_out(const float* __restrict__ X5,
                                             float* __restrict__ out, int n) {
  const int g = blockIdx.x * 256 + threadIdx.x;
  if (g < n) out[g] = X5[g];
}

// ---------------------------------------------------------------------------
extern "C" void kernel_launch(void* const* d_in, const int* in_sizes, int n_in,
                              void* d_out, int out_size, void* d_ws, size_t ws_size,
                              hipStream_t stream) {
  const float* obs = (const float*)d_in[0];
  const float* x1i = (const float*)d_in[1];
  const float* x2i = (const float*)d_in[2];
  const float* x3i = (const float*)d_in[3];
  const float* x4i = (const float*)d_in[4];
  const float* x5i = (const float*)d_in[5];
  const float* W1g = (const float*)d_in[6];
  const float* b1g = (const float*)d_in[7];
  const float* W2g = (const float*)d_in[8];
  const float* b2g = (const float*)d_in[9];
  const float* W3g = (const float*)d_in[10];
  const float* b3g = (const float*)d_in[11];
  const float* W4g = (const float*)d_in[12];
  const float* b4g = (const float*)d_in[13];
  const float* W5g = (const float*)d_in[14];
  const float* b5g = (const float*)d_in[15];
  const int B = in_sizes[0] / 784;  // 2048

  float* ws = (float*)d_ws;
  size_t o = 0;
  float* X1 = ws + o; o += (size_t)B * 18432;
  float* X2 = ws + o; o += (size_t)B * 6400;
  float* X3 = ws + o; o += (size_t)B * 576;
  float* X4 = ws + o; o += (size_t)B * 128;
  float* X5 = ws + o; o += (size_t)B * 10;
  float* E0 = ws + o; o += (size_t)B * 784;
  float* E1 = ws + o; o += (size_t)B * 18432;
  float* E2 = ws + o; o += (size_t)B * 6400;
  float* E3 = ws + o; o += (size_t)B * 576;
  float* E4 = ws + o; o += (size_t)B * 128;
  float* K2 = ws + o; o += 64 * 1152;
  float* K3 = ws + o; o += 64 * 2304;
  (void)o; (void)ws_size; (void)n_in; (void)out_size;

  k_init<<<dim3(2048), dim3(256), 0, stream>>>(x1i, x2i, x3i, x4i, x5i, W2g, W3g,
                                               X1, X2, X3, X4, X5, K2, K3, B);
  const int e4blocks = (B * 128 + 255) / 256;
  const int x5blocks = (B * 10 + 255) / 256;
  for (int step = 0; step < STEPS; ++step) {
    k_e0e4<<<dim3(B + e4blocks), dim3(256), 0, stream>>>(obs, W1g, b1g, W5g, b5g,
                                                         X1, X4, X5, E0, E4, B);
    k_e1<<<dim3(B), dim3(256), 0, stream>>>(W2g, b2g, X1, X2, E1);
    k_e2<<<dim3(B), dim3(256), 0, stream>>>(W3g, b3g, X2, X3, E2);
    k_e3<<<dim3(B / 16), dim3(256), 0, stream>>>(W4g, b4g, X3, X4, E3);
    k_x1x5<<<dim3(B + x5blocks), dim3(256), 0, stream>>>(W1g, W5g, E0, E1, E4,
                                                         X1, X5, B);
    k_x2<<<dim3(B), dim3(256), 0, stream>>>(K2, E1, E2, X2);
    k_x3<<<dim3(B / 4), dim3(256), 0, stream>>>(K3, E2, E3, X3);
    k_x4<<<dim3(B / 16), dim3(256), 0, stream>>>(W4g, E3, E4, X4);
  }
  k_out<<<dim3(x5blocks), dim3(256), 0, stream>>>(X5, (float*)d_out, B * 10);
}